// PPoly_45165876085464
// MI455X (gfx1250) — compile-verified
//
#include <hip/hip_runtime.h>

// PPoly cubic evaluation: y[q, d] = ((c0*t + c1)*t + c2)*t + c3 on interval i(q).
// Pure bandwidth kernel (64 MB streaming store dominates); no matrix-op shape
// exists (coefficients are gathered per query). Optimized for gfx1250 wave32:
// 16 lanes per query, 128-bit coalesced vmem, non-temporal output stores.

typedef float v4f __attribute__((ext_vector_type(4)));

namespace {
constexpr int kNIntervals = 4096;
constexpr int kD          = 64;
constexpr int kLanesPerQ  = 16;                    // each lane: one float4 of 64 outputs
constexpr int kBlock      = 256;                   // 8 wave32s
constexpr int kQPerBlock  = kBlock / kLanesPerQ;   // 16 queries per block
}

__global__ __launch_bounds__(kBlock) void ppoly_eval_kernel(
    const float* __restrict__ xq,
    const float* __restrict__ c,     // [4, 4096, 64]
    const float* __restrict__ xb,    // [4097]
    float* __restrict__ out,         // [nq, 64]
    int nq)
{
  const int tid = threadIdx.x;
  const int q   = blockIdx.x * kQPerBlock + (tid >> 4);
  const int dg  = tid & (kLanesPerQ - 1);          // which float4 of the row
  if (q >= nq) return;

  const float x = xq[q];                           // broadcast across 16 lanes

  // xb = linspace(-4, 4, 4097): step h = 2^-9 exactly. Analytic guess for
  // i = clip(searchsorted(xb, x, side='right'), 1, 4096):
  int i = (int)floorf((x + 4.0f) * 512.0f) + 1;
  i = i < 1 ? 1 : (i > kNIntervals ? kNIntervals : i);
  // Fixup vs the actual breakpoint array: handles the rare half-ulp rounding
  // of (x+4) across a breakpoint; xb sorted -> terminates (~0 iterations).
  while (i > 1 && x < xb[i - 1]) --i;
  while (i < kNIntervals && x >= xb[i]) ++i;

  const float t = x - xb[i - 1];

  // Gather the 4x64 coefficient slab: per query, 4 x 256B coalesced bursts
  // (global_load_b128 per lane). c rows are 1 MB apart; table is L2-resident.
  const v4f* __restrict__ cv = (const v4f*)c;
  const size_t base = (size_t)(i - 1) * (kD / 4) + (size_t)dg;
  const size_t mstr = (size_t)kNIntervals * (kD / 4);
  const v4f c0 = cv[base];
  const v4f c1 = cv[base + mstr];
  const v4f c2 = cv[base + 2 * mstr];
  const v4f c3 = cv[base + 3 * mstr];

  v4f y;
  y.x = fmaf(fmaf(fmaf(c0.x, t, c1.x), t, c2.x), t, c3.x);
  y.y = fmaf(fmaf(fmaf(c0.y, t, c1.y), t, c2.y), t, c3.y);
  y.z = fmaf(fmaf(fmaf(c0.z, t, c1.z), t, c2.z), t, c3.z);
  y.w = fmaf(fmaf(fmaf(c0.w, t, c1.w), t, c2.w), t, c3.w);

  // 64 MB write-once stream: non-temporal hint (TH=NT) keeps c/xb hot in L2.
  v4f* __restrict__ ov = (v4f*)out;
  __builtin_nontemporal_store(y, &ov[(size_t)q * (kD / 4) + dg]);
}

extern "C" void kernel_launch(void* const* d_in, const int* in_sizes, int n_in,
                              void* d_out, int out_size, void* d_ws, size_t ws_size,
                              hipStream_t stream) {
  (void)n_in; (void)out_size; (void)d_ws; (void)ws_size;
  const float* xq = (const float*)d_in[0];   // [N_QUERY]
  const float* c  = (const float*)d_in[1];   // [4, 4096, 64]
  const float* xb = (const float*)d_in[2];   // [4097]
  float* out      = (float*)d_out;           // [N_QUERY, 64]

  const int nq     = in_sizes[0];
  const int blocks = (nq + kQPerBlock - 1) / kQPerBlock;
  ppoly_eval_kernel<<<blocks, kBlock, 0, stream>>>(xq, c, xb, out, nq);
}